// HeteroDotProductPredictor_25855703122642
// MI455X (gfx1250) — compile-verified
//
#include <hip/hip_runtime.h>
#include <hip/hip_bf16.h>

typedef __attribute__((ext_vector_type(2))) float v2f;
typedef __attribute__((ext_vector_type(4))) float v4f;
typedef __attribute__((ext_vector_type(8))) float v8f;

#define WAVES_PER_BLOCK 8
#define DFEAT 128

// One wave32 computes 16 edge scores.
//   Lane m   (m<16) : accumulates  sum_{k=0..63}  h[src[m],k]*h[dst[m],k]  (2-wide pk_fma)
//   Lane m+16       : accumulates  sum_{k=64..127} ...
// A single V_WMMA_F32_16X16X4_F32 with an all-ones B then reduces the 4 per-row
// partials (2 per lane x 2 half-wave lanes) across lanes: D[m,n] = score[m] for all n.
// This uses the matrix pipe as the cross-lane reduction + store-transpose in one op.
__global__ __launch_bounds__(256) void
edge_dot_wmma_kernel(const float* __restrict__ h,
                     const int* __restrict__ src_idx,
                     const int* __restrict__ dst_idx,
                     float* __restrict__ out,
                     int n_edges, int n_tiles) {
    const int lane = threadIdx.x & 31;
    const int wave = threadIdx.x >> 5;
    const int tile = blockIdx.x * WAVES_PER_BLOCK + wave;
    if (tile >= n_tiles) return;          // whole-wave exit; EXEC stays all-ones below

    const int m    = lane & 15;           // edge row within tile (A-matrix M)
    const int koff = (lane >> 4) << 6;    // lanes 16-31 own the upper 64 features

    int e = tile * 16 + m;
    if (e >= n_edges) e = n_edges - 1;    // clamp instead of masking: keep EXEC full

    const int s = src_idx[e];
    const int d = dst_idx[e];
    const float* __restrict__ pu = h + (size_t)s * DFEAT + koff;
    const float* __restrict__ pv = h + (size_t)d * DFEAT + koff;

    v2f acc2 = {0.0f, 0.0f};

#pragma unroll 8
    for (int k = 0; k < DFEAT / 2; k += 4) {
        v4f au = *(const v4f*)(pu + k);   // 16B gather chunk of src row (b128)
        v4f av = *(const v4f*)(pv + k);   // 16B gather chunk of dst row (b128)
        v2f ulo = {au[0], au[1]}, uhi = {au[2], au[3]};
        v2f vlo = {av[0], av[1]}, vhi = {av[2], av[3]};
        acc2 = ulo * vlo + acc2;          // v_pk_fma_f32
        acc2 = uhi * vhi + acc2;          // v_pk_fma_f32
    }

    // Cross-lane reduce via the matrix pipe: A(16x4) = per-lane partials,
    // B(4x16) = ones  ->  D[m,n] = full dot product of edge m in every column.
    const v2f ones = {1.0f, 1.0f};
    v8f czero = {};
    v8f acc = __builtin_amdgcn_wmma_f32_16x16x4_f32(
                  false, acc2, false, ones, (short)0, czero, false, false);

    // C/D layout: lane 0 holds M=0..7 (N=0) in acc[0..7]; lane 16 holds M=8..15.
    if ((lane & 15) == 0) {
        const int base = tile * 16 + ((lane >> 4) << 3);
        if (base + 8 <= n_edges) {
            v4f lo = {acc[0], acc[1], acc[2], acc[3]};
            v4f hi = {acc[4], acc[5], acc[6], acc[7]};
            *(v4f*)(out + base)     = lo;   // 16B-aligned (base % 8 == 0)
            *(v4f*)(out + base + 4) = hi;
        } else {
            float vals[8] = {acc[0], acc[1], acc[2], acc[3],
                             acc[4], acc[5], acc[6], acc[7]};
            for (int r = 0; r < 8; ++r)
                if (base + r < n_edges) out[base + r] = vals[r];
        }
    }
}

extern "C" void kernel_launch(void* const* d_in, const int* in_sizes, int n_in,
                              void* d_out, int out_size, void* d_ws, size_t ws_size,
                              hipStream_t stream) {
    const float* h       = (const float*)d_in[0];
    const int*   src_idx = (const int*)d_in[1];
    const int*   dst_idx = (const int*)d_in[2];
    float*       out     = (float*)d_out;

    const int n_edges = in_sizes[1];                 // 640000
    const int n_tiles = (n_edges + 15) / 16;         // 16 edges per wave
    const int blocks  = (n_tiles + WAVES_PER_BLOCK - 1) / WAVES_PER_BLOCK;

    edge_dot_wmma_kernel<<<blocks, 256, 0, stream>>>(h, src_idx, dst_idx, out,
                                                     n_edges, n_tiles);
}